// DownSampler_74096775790725
// MI455X (gfx1250) — compile-verified
//
#include <hip/hip_runtime.h>
#include <hip/hip_bf16.h>

#define BATCH   16
#define NNODE   16384
#define NCKEEP  4096
#define DIN     128
#define DOUT    128
#define KNN     6

typedef __attribute__((ext_vector_type(2))) float v2f;
typedef __attribute__((ext_vector_type(8))) float v8f;

// ---------------------------------------------------------------------------
// gfx1250 async memory->LDS helpers (ASYNCcnt-tracked; bypass VGPR round-trip)
// ---------------------------------------------------------------------------
__device__ __forceinline__ unsigned lds_off_u32(const void* p) {
  // flat LDS address = {aperture_hi, lds_byte_offset}; hardware uses addr[31:0]
  return (unsigned)(unsigned long long)(uintptr_t)p;
}
__device__ __forceinline__ void async_load_b128(unsigned lds_off, const void* g) {
  asm volatile("global_load_async_to_lds_b128 %0, %1, off"
               :: "v"(lds_off), "v"((unsigned long long)(uintptr_t)g)
               : "memory");
}
__device__ __forceinline__ void wait_async0() {
#if __has_builtin(__builtin_amdgcn_s_wait_asynccnt)
  __builtin_amdgcn_s_wait_asynccnt(0);
#else
  asm volatile("s_wait_asynccnt 0" ::: "memory");
#endif
}

// ---------------------------------------------------------------------------
// K1: score[row] = dot(x[row,:], w_sel)   (softmax / norm skipped: monotonic)
// ---------------------------------------------------------------------------
__global__ __launch_bounds__(256) void score_kernel(
    const float* __restrict__ x, const float* __restrict__ w_sel,
    float* __restrict__ score) {
  int wave = (blockIdx.x * blockDim.x + threadIdx.x) >> 5;
  int lane = threadIdx.x & 31;
  if (wave >= BATCH * NNODE) return;
  const float4 a = ((const float4*)(x + (size_t)wave * DIN))[lane];
  const float4 w = ((const float4*)w_sel)[lane];
  float s = a.x * w.x + a.y * w.y + a.z * w.z + a.w * w.w;
  #pragma unroll
  for (int off = 16; off > 0; off >>= 1) s += __shfl_xor(s, off);
  if (lane == 0) score[wave] = s;
}

// ---------------------------------------------------------------------------
// K2: per-graph full bitonic sort of 16384 (score,idx) keys in 128KB LDS.
// key = sortable_uint(score)<<32 | ~idx  (descending sort == jax ordering)
// ---------------------------------------------------------------------------
__global__ __launch_bounds__(1024) void topk_sort_kernel(
    const float* __restrict__ score, int* __restrict__ perm) {
  extern __shared__ unsigned long long skey[];   // 16384 * 8B = 128KB dynamic
  const int b = blockIdx.x;
  const int t = threadIdx.x;

  for (int i = t; i < NNODE; i += 1024) {
    unsigned u = __float_as_uint(score[b * NNODE + i]);
    u = (u & 0x80000000u) ? ~u : (u | 0x80000000u);
    skey[i] = ((unsigned long long)u << 32) | (unsigned)(~i);
  }
  __syncthreads();

  for (int ksz = 2; ksz <= NNODE; ksz <<= 1) {
    for (int j = ksz >> 1; j > 0; j >>= 1) {
      for (int e = t; e < NNODE / 2; e += 1024) {
        int i   = ((e & ~(j - 1)) << 1) | (e & (j - 1));
        int ixj = i | j;
        unsigned long long a = skey[i], c = skey[ixj];
        bool desc = ((i & ksz) == 0);
        bool sw = desc ? (a < c) : (a > c);
        if (sw) { skey[i] = c; skey[ixj] = a; }
      }
      __syncthreads();
    }
  }

  for (int r = t; r < NCKEEP; r += 1024) {
    int local = (int)(~(unsigned)skey[r]);
    perm[b * NCKEEP + r] = b * NNODE + local;
  }
}

// ---------------------------------------------------------------------------
// K3: x_c = x[perm] @ w_lin^T + b_lin via V_WMMA_F32_16X16X4_F32.
// Block = 256 thr = 8 waves, M-tile = 64 rows. Wave w: rows [16*(w>>1), +16),
// cols [64*(w&1), +64) as 4 16x16 subtiles -> 4 v8f accumulators, 128 WMMAs.
// W (64KB) + A tile (32KB) staged into LDS via global_load_async_to_lds_b128.
// ---------------------------------------------------------------------------
__global__ __launch_bounds__(256) void gather_linear_kernel(
    const float* __restrict__ x, const int* __restrict__ perm,
    const float* __restrict__ w_lin, const float* __restrict__ b_lin,
    float* __restrict__ x_c) {
  extern __shared__ float fsm[];                 // [128*128 W | 64*128 A]
  float* sW = fsm;
  float* sA = fsm + DOUT * DIN;
  const int t = threadIdx.x;
  const int m0 = blockIdx.x * 64;

  // async-stage W: 16384 floats = 4096 x b128, 16 per thread
  #pragma unroll
  for (int i = 0; i < 16; ++i) {
    int idx = i * 256 + t;
    async_load_b128(lds_off_u32(sW + idx * 4), w_lin + idx * 4);
  }
  // async-stage gathered A: 64 rows x 128 = 2048 x b128, 8 per thread
  #pragma unroll
  for (int i = 0; i < 8; ++i) {
    int idx = i * 256 + t;                       // (row, col4): 32 b128 per row
    int r = idx >> 5;
    int c = idx & 31;
    int g = perm[m0 + r];
    async_load_b128(lds_off_u32(sA + idx * 4), x + (size_t)g * DIN + c * 4);
  }
  wait_async0();
  __syncthreads();

  const int lane = t & 31, wave = t >> 5;
  const int lm = lane & 15, hi = lane >> 4;
  const int mi = wave >> 1;                      // 0..3 : row subtile
  const int nbase = (wave & 1) * 64;             // 0 or 64 : col half
  const float* arow = sA + (16 * mi + lm) * DIN;
  const float* br0 = sW + (nbase + 0  + lm) * DIN;
  const float* br1 = sW + (nbase + 16 + lm) * DIN;
  const float* br2 = sW + (nbase + 32 + lm) * DIN;
  const float* br3 = sW + (nbase + 48 + lm) * DIN;

  v8f acc0 = {0.f,0.f,0.f,0.f,0.f,0.f,0.f,0.f};
  v8f acc1 = acc0, acc2 = acc0, acc3 = acc0;

  #pragma unroll
  for (int kk = 0; kk < DIN / 4; ++kk) {
    int k0 = kk * 4 + 2 * hi;
    v2f a  = *(const v2f*)(arow + k0);
    v2f b0 = *(const v2f*)(br0 + k0);
    v2f b1 = *(const v2f*)(br1 + k0);
    v2f b2 = *(const v2f*)(br2 + k0);
    v2f b3 = *(const v2f*)(br3 + k0);
    acc0 = __builtin_amdgcn_wmma_f32_16x16x4_f32(false, a, false, b0, (short)0, acc0, false, false);
    acc1 = __builtin_amdgcn_wmma_f32_16x16x4_f32(false, a, false, b1, (short)0, acc1, false, false);
    acc2 = __builtin_amdgcn_wmma_f32_16x16x4_f32(false, a, false, b2, (short)0, acc2, false, false);
    acc3 = __builtin_amdgcn_wmma_f32_16x16x4_f32(false, a, false, b3, (short)0, acc3, false, false);
  }

  const int mrow = m0 + 16 * mi + 8 * hi;        // + r below
  #pragma unroll
  for (int c = 0; c < 4; ++c) {
    int n = nbase + 16 * c + lm;
    float bias = b_lin[n];
    v8f acc = (c == 0) ? acc0 : (c == 1) ? acc1 : (c == 2) ? acc2 : acc3;
    #pragma unroll
    for (int r = 0; r < 8; ++r) {
      x_c[(size_t)(mrow + r) * DOUT + n] = acc[r] + bias;
    }
  }
}

// ---------------------------------------------------------------------------
// K3b: pos_c / batch_c gather
// ---------------------------------------------------------------------------
__global__ __launch_bounds__(256) void gather_pos_kernel(
    const float* __restrict__ pos, const int* __restrict__ perm,
    float* __restrict__ pos_c, int* __restrict__ batch_c) {
  int i = blockIdx.x * blockDim.x + threadIdx.x;
  if (i >= BATCH * NCKEEP) return;
  int g = perm[i];
  pos_c[i * 3 + 0] = pos[(size_t)g * 3 + 0];
  pos_c[i * 3 + 1] = pos[(size_t)g * 3 + 1];
  pos_c[i * 3 + 2] = pos[(size_t)g * 3 + 2];
  batch_c[i] = i / NCKEEP;
}

// ---------------------------------------------------------------------------
// K4: KNN (k=6, no self loop) per graph over pos_c; register top-6.
// ---------------------------------------------------------------------------
__global__ __launch_bounds__(256) void knn_kernel(
    const float* __restrict__ pos_c, int* __restrict__ ei) {
  __shared__ float sx[256], sy[256], sz[256];
  const int blocks_per_graph = NCKEEP / 256;
  const int b = blockIdx.x / blocks_per_graph;
  const int i_local = (blockIdx.x % blocks_per_graph) * 256 + threadIdx.x;
  const int i = b * NCKEEP + i_local;

  const float px = pos_c[(size_t)i * 3 + 0];
  const float py = pos_c[(size_t)i * 3 + 1];
  const float pz = pos_c[(size_t)i * 3 + 2];

  float bd[KNN]; int bi[KNN];
  #pragma unroll
  for (int k = 0; k < KNN; ++k) { bd[k] = __builtin_inff(); bi[k] = 0; }

  for (int tile = 0; tile < NCKEEP / 256; ++tile) {
    __syncthreads();
    int j0 = tile * 256;
    int jg = b * NCKEEP + j0 + threadIdx.x;
    sx[threadIdx.x] = pos_c[(size_t)jg * 3 + 0];
    sy[threadIdx.x] = pos_c[(size_t)jg * 3 + 1];
    sz[threadIdx.x] = pos_c[(size_t)jg * 3 + 2];
    if (tile + 1 < NCKEEP / 256) {               // gfx1250 global_prefetch_b8
      __builtin_prefetch(pos_c + (size_t)(jg + 256) * 3, 0, 1);
    }
    __syncthreads();

    #pragma unroll 4
    for (int jj = 0; jj < 256; ++jj) {
      float dx = px - sx[jj];
      float dy = py - sy[jj];
      float dz = pz - sz[jj];
      float d2 = dx * dx + dy * dy + dz * dz;
      int jl = j0 + jj;
      if (jl != i_local && d2 < bd[KNN - 1]) {
        float d = d2; int id = jl;
        #pragma unroll
        for (int k = 0; k < KNN; ++k) {
          bool c = d < bd[k];
          float td = bd[k]; int ti = bi[k];
          bd[k] = c ? d : bd[k];  bi[k] = c ? id : bi[k];
          d = c ? td : d;         id = c ? ti : id;
        }
      }
    }
  }

  const int E = BATCH * NCKEEP * KNN;
  int base = i * KNN;
  #pragma unroll
  for (int k = 0; k < KNN; ++k) {
    int src = b * NCKEEP + bi[k];
    int ctr = i;
    ei[base + k]         = src;
    ei[E + base + k]     = ctr;
    ei[2 * E + base + k] = ctr;
    ei[3 * E + base + k] = src;
  }
}

// ---------------------------------------------------------------------------
// Launch
// ---------------------------------------------------------------------------
extern "C" void kernel_launch(void* const* d_in, const int* in_sizes, int n_in,
                              void* d_out, int out_size, void* d_ws, size_t ws_size,
                              hipStream_t stream) {
  const float* x     = (const float*)d_in[0];
  const float* pos   = (const float*)d_in[1];
  // d_in[2] = batch (recomputed analytically)
  const float* w_sel = (const float*)d_in[3];
  const float* w_lin = (const float*)d_in[4];
  const float* b_lin = (const float*)d_in[5];

  float* score = (float*)d_ws;                                    // B*N floats
  int*   perm  = (int*)((char*)d_ws + (size_t)BATCH * NNODE * 4); // B*NC ints

  float* out    = (float*)d_out;
  float* x_c    = out;                                             // 8388608
  float* pos_c  = out + (size_t)BATCH * NCKEEP * DOUT;             // 196608
  int*   batchc = (int*)(pos_c + (size_t)BATCH * NCKEEP * 3);      // 65536
  int*   ei     = (int*)(batchc + (size_t)BATCH * NCKEEP);         // 4*E ints

  score_kernel<<<(BATCH * NNODE) / 8, 256, 0, stream>>>(x, w_sel, score);

  topk_sort_kernel<<<BATCH, 1024, NNODE * sizeof(unsigned long long), stream>>>(
      score, perm);

  gather_linear_kernel<<<(BATCH * NCKEEP) / 64, 256,
                         (DOUT * DIN + 64 * DIN) * sizeof(float), stream>>>(
      x, perm, w_lin, b_lin, x_c);

  gather_pos_kernel<<<(BATCH * NCKEEP + 255) / 256, 256, 0, stream>>>(
      pos, perm, pos_c, batchc);

  knn_kernel<<<BATCH * (NCKEEP / 256), 256, 0, stream>>>(pos_c, ei);
}